// SparseGlobalBlock_37512244363868
// MI455X (gfx1250) — compile-verified
//
#include <hip/hip_runtime.h>

// ---------------------------------------------------------------------------
// Sparse global attention block for MI455X (gfx1250, wave32, WMMA).
// GEMMs: V_WMMA_F32_16X16X4_F32 with LDS tile staging via
// GLOBAL_LOAD_ASYNC_TO_LDS_B128 (ASYNCcnt-tracked CDNA5 async copy).
// ---------------------------------------------------------------------------

#define BB 16
#define CC 256
#define HH 64
#define WW 64
#define NN (HH * WW)      // 4096 tokens
#define KT 512            // top-k tokens
#define LN_EPS 1e-5f
#define CLAMPV 80.0f
#define ATTN_SCALE 0.0625f  // 256^-0.5

#define KSTEP 32
#define LDS_MS 36   // padded stride (floats) for m/n-major tiles: 144B, 16B-aligned
#define LDS_KS 68   // padded stride (floats) for k-major B tile: 272B, 16B-aligned

typedef __attribute__((ext_vector_type(2))) float v2f;
typedef __attribute__((ext_vector_type(8))) float v8f;

__device__ __forceinline__ float nan_to_zero(float v) {
    if (!(v == v) || fabsf(v) > 3.0e38f) return 0.f;
    return v;
}

// CDNA5 async global->LDS copy (GV mode): vdst = LDS byte offset, vaddr = 64b.
// Generic pointers to __shared__ carry the LDS offset in their low 32 bits.
__device__ __forceinline__ void async_copy_b128(const float* lds_dst, const float* gsrc) {
    unsigned lds_off = (unsigned)(unsigned long long)lds_dst;
    asm volatile("global_load_async_to_lds_b128 %0, %1, off"
                 :: "v"(lds_off), "v"(gsrc) : "memory");
}
__device__ __forceinline__ void wait_async_all() {
    asm volatile("s_wait_asynccnt 0x0" ::: "memory");
}

// ---------------------------------------------------------------------------
// 1) importance[b][n] = nan_to_num( sum_c x[b][c][n]^2 )
// ---------------------------------------------------------------------------
__global__ __launch_bounds__(256)
void importance_kernel(const float* __restrict__ x, float* __restrict__ imp) {
    int flat = blockIdx.x * 256 + threadIdx.x;     // over B*N
    int b = flat >> 12;
    int n = flat & (NN - 1);
    const float* xp = x + (size_t)b * CC * NN + n;
    float s = 0.f;
    for (int c = 0; c < CC; ++c) {
        float v = xp[(size_t)c * NN];
        s += v * v;
    }
    imp[flat] = nan_to_zero(s);
}

// ---------------------------------------------------------------------------
// 2) per-batch top-512 of 4096 via bitonic sort in LDS (32 KB).
// ---------------------------------------------------------------------------
__global__ __launch_bounds__(1024)
void topk_kernel(const float* __restrict__ imp, int* __restrict__ idxOut) {
    __shared__ unsigned long long s[NN];
    const int b = blockIdx.x;
    const int tid = threadIdx.x;
    for (int i = tid; i < NN; i += 1024) {
        float f = imp[b * NN + i];
        unsigned u = __float_as_uint(f);
        u = (u & 0x80000000u) ? ~u : (u | 0x80000000u);  // monotone map
        s[i] = ((unsigned long long)(~u) << 32) | (unsigned)i;
    }
    __syncthreads();
    for (int k = 2; k <= NN; k <<= 1) {
        for (int j = k >> 1; j > 0; j >>= 1) {
            for (int i = tid; i < NN; i += 1024) {
                int ixj = i ^ j;
                if (ixj > i) {
                    bool up = ((i & k) == 0);
                    unsigned long long a = s[i], c = s[ixj];
                    if ((a > c) == up) { s[i] = c; s[ixj] = a; }
                }
            }
            __syncthreads();
        }
    }
    if (tid < KT) idxOut[b * KT + tid] = (int)(s[tid] & 0xFFFFFFFFull);
}

// ---------------------------------------------------------------------------
// 3) gather: dst[b][t][c] = src[b][c][idx[b][t]]   (src is (B,C,N))
// ---------------------------------------------------------------------------
__global__ __launch_bounds__(256)
void gather_tokens(const float* __restrict__ src, const int* __restrict__ idx,
                   float* __restrict__ dst) {
    int flat = blockIdx.x * 256 + threadIdx.x;     // over B*KT*CC
    int c = flat & (CC - 1);
    int t = (flat >> 8) & (KT - 1);
    int b = flat >> 17;
    int n = idx[b * KT + t];
    dst[flat] = src[(size_t)b * CC * NN + (size_t)c * NN + n];
}

__global__ __launch_bounds__(256)
void scatter_tokens(const int* __restrict__ idx, const float* __restrict__ att,
                    float* __restrict__ v) {
    int flat = blockIdx.x * 256 + threadIdx.x;
    int c = flat & (CC - 1);
    int t = (flat >> 8) & (KT - 1);
    int b = flat >> 17;
    int n = idx[b * KT + t];
    v[(size_t)b * CC * NN + (size_t)c * NN + n] = att[flat];
}

// ---------------------------------------------------------------------------
// 4) WMMA f32 GEMM with async-LDS staging.
//    D(MxN) = A(MxK row-major) * B.  TRANSB=0: B (KxN) row-major.
//    TRANSB=1: B(k,n) = B[n*ldb + k] (stored NxK row-major).
//    Block = 256 thr / 8 waves; block tile 64x64; wave tile 16x32 (2 accs).
//    K-step 32: A 64x32 + B tile staged via GLOBAL_LOAD_ASYNC_TO_LDS_B128.
//    EPI: 0=store, 1=attn pre-softmax (scale, nan->0, clamp), 2=+residual.
// ---------------------------------------------------------------------------
template <bool TRANSB, int EPI>
__global__ __launch_bounds__(256)
void gemm_wmma_f32(const float* __restrict__ Ag, const float* __restrict__ Bg,
                   float* __restrict__ Dg, const float* __restrict__ Rg,
                   int K, int lda, int ldb, int ldd,
                   long long strA, long long strB, long long strD, long long strR) {
    __shared__ float sA[64 * LDS_MS];             // 9216 B
    __shared__ float sB[64 * LDS_MS];             // covers 32*LDS_KS too

    const int z = blockIdx.z;
    const float* A = Ag + strA * z;
    const float* Bm = Bg + strB * z;
    float* D = Dg + strD * z;
    const float* R = (EPI == 2) ? (Rg + strR * z) : nullptr;

    const int tid = threadIdx.x;
    const int lane = tid & 31;
    const int wave = tid >> 5;
    const int hl = lane >> 4;          // lane half: K sub-pair / M+8 rows
    const int lm = lane & 15;
    const int wm = wave >> 1;          // 0..3  (M sub-tile)
    const int wn = wave & 1;           // 0..1  (N sub-tile pair)
    const int blockM = blockIdx.y * 64;
    const int blockN = blockIdx.x * 64;
    const int tileM = blockM + wm * 16;
    const int tileN0 = blockN + wn * 32;

    // staging geometry (each thread: 2 x B128 per tile)
    const int arow = tid >> 3;         // 0..31 (+32)
    const int aseg = tid & 7;          // 8 segs x 4 floats = 32 K
    const int brow = tid >> 4;         // 0..15 (+16)   (k-major B)
    const int bseg = tid & 15;         // 16 segs x 4 floats = 64 N

    v8f acc0 = {0.f, 0.f, 0.f, 0.f, 0.f, 0.f, 0.f, 0.f};
    v8f acc1 = {0.f, 0.f, 0.f, 0.f, 0.f, 0.f, 0.f, 0.f};

    for (int k0 = 0; k0 < K; k0 += KSTEP) {
        __syncthreads();   // previous iteration's fragment reads done
        // ---- stage A tile (64 x 32, m-major, stride LDS_MS) ----
        async_copy_b128(&sA[arow * LDS_MS + aseg * 4],
                        A + (size_t)(blockM + arow) * lda + k0 + aseg * 4);
        async_copy_b128(&sA[(arow + 32) * LDS_MS + aseg * 4],
                        A + (size_t)(blockM + arow + 32) * lda + k0 + aseg * 4);
        if (TRANSB) {
            // ---- B tile: 64 n-rows x 32 k (n-major, stride LDS_MS) ----
            async_copy_b128(&sB[arow * LDS_MS + aseg * 4],
                            Bm + (size_t)(blockN + arow) * ldb + k0 + aseg * 4);
            async_copy_b128(&sB[(arow + 32) * LDS_MS + aseg * 4],
                            Bm + (size_t)(blockN + arow + 32) * ldb + k0 + aseg * 4);
        } else {
            // ---- B tile: 32 k-rows x 64 n (k-major, stride LDS_KS) ----
            async_copy_b128(&sB[brow * LDS_KS + bseg * 4],
                            Bm + (size_t)(k0 + brow) * ldb + blockN + bseg * 4);
            async_copy_b128(&sB[(brow + 16) * LDS_KS + bseg * 4],
                            Bm + (size_t)(k0 + brow + 16) * ldb + blockN + bseg * 4);
        }
        wait_async_all();  // this wave's async copies landed in LDS
        __syncthreads();   // all waves' tiles visible

#pragma unroll
        for (int kk = 0; kk < KSTEP; kk += 4) {
            v2f a;
            const float* ap = &sA[(wm * 16 + lm) * LDS_MS + kk + 2 * hl];
            a.x = ap[0];
            a.y = ap[1];
            v2f b0, b1;
            if (TRANSB) {
                const float* bp0 = &sB[(wn * 32 + lm) * LDS_MS + kk + 2 * hl];
                const float* bp1 = &sB[(wn * 32 + 16 + lm) * LDS_MS + kk + 2 * hl];
                b0.x = bp0[0]; b0.y = bp0[1];
                b1.x = bp1[0]; b1.y = bp1[1];
            } else {
                const float* bp = &sB[(kk + 2 * hl) * LDS_KS + wn * 32 + lm];
                b0.x = bp[0];       b0.y = bp[LDS_KS];
                b1.x = bp[16];      b1.y = bp[LDS_KS + 16];
            }
            acc0 = __builtin_amdgcn_wmma_f32_16x16x4_f32(
                false, a, false, b0, (short)0, acc0, false, false);
            acc1 = __builtin_amdgcn_wmma_f32_16x16x4_f32(
                false, a, false, b1, (short)0, acc1, false, false);
        }
    }

#pragma unroll
    for (int r = 0; r < 8; ++r) {
        int row = tileM + r + 8 * hl;
        int c0 = tileN0 + lm;
        int c1 = tileN0 + 16 + lm;
        float v0 = acc0[r];
        float v1 = acc1[r];
        if (EPI == 1) {
            v0 = fminf(fmaxf(nan_to_zero(v0 * ATTN_SCALE), -CLAMPV), CLAMPV);
            v1 = fminf(fmaxf(nan_to_zero(v1 * ATTN_SCALE), -CLAMPV), CLAMPV);
        } else if (EPI == 2) {
            v0 += R[(size_t)row * ldd + c0];
            v1 += R[(size_t)row * ldd + c1];
        }
        D[(size_t)row * ldd + c0] = v0;
        D[(size_t)row * ldd + c1] = v1;
    }
}

// ---------------------------------------------------------------------------
// 5) LayerNorm over C=256 per (b, token) row; one block per row.
// ---------------------------------------------------------------------------
__global__ __launch_bounds__(256)
void layernorm_rows(float* __restrict__ buf, const float* __restrict__ w,
                    const float* __restrict__ bias) {
    __shared__ float red[256];
    const int tid = threadIdx.x;
    float* p = buf + (size_t)blockIdx.x * CC;
    float v = p[tid];
    red[tid] = v;
    __syncthreads();
    for (int s = 128; s > 0; s >>= 1) {
        if (tid < s) red[tid] += red[tid + s];
        __syncthreads();
    }
    float mu = red[0] * (1.0f / CC);
    __syncthreads();
    float d = v - mu;
    red[tid] = d * d;
    __syncthreads();
    for (int s = 128; s > 0; s >>= 1) {
        if (tid < s) red[tid] += red[tid + s];
        __syncthreads();
    }
    float var = red[0] * (1.0f / CC);
    p[tid] = d * rsqrtf(var + LN_EPS) * w[tid] + bias[tid];
}

// ---------------------------------------------------------------------------
// 6) softmax over rows of length 512; one block (256 thr, 2 elems/thr).
// ---------------------------------------------------------------------------
__global__ __launch_bounds__(256)
void softmax_rows(float* __restrict__ attn) {
    __shared__ float red[256];
    const int tid = threadIdx.x;
    float* p = attn + (size_t)blockIdx.x * KT;
    float a0 = p[tid], a1 = p[tid + 256];
    red[tid] = fmaxf(a0, a1);
    __syncthreads();
    for (int s = 128; s > 0; s >>= 1) {
        if (tid < s) red[tid] = fmaxf(red[tid], red[tid + s]);
        __syncthreads();
    }
    float m = red[0];
    __syncthreads();
    float e0 = __expf(a0 - m), e1 = __expf(a1 - m);
    red[tid] = e0 + e1;
    __syncthreads();
    for (int s = 128; s > 0; s >>= 1) {
        if (tid < s) red[tid] += red[tid + s];
        __syncthreads();
    }
    float inv = 1.0f / red[0];
    p[tid]       = nan_to_zero(e0 * inv);
    p[tid + 256] = nan_to_zero(e1 * inv);
}

// ---------------------------------------------------------------------------
// Launch
// ---------------------------------------------------------------------------
extern "C" void kernel_launch(void* const* d_in, const int* in_sizes, int n_in,
                              void* d_out, int out_size, void* d_ws, size_t ws_size,
                              hipStream_t stream) {
    (void)in_sizes; (void)n_in; (void)out_size; (void)ws_size;
    const float* x   = (const float*)d_in[0];
    const float* wq  = (const float*)d_in[1];
    const float* wk  = (const float*)d_in[2];
    const float* wv  = (const float*)d_in[3];
    const float* wo  = (const float*)d_in[4];
    const float* lnw = (const float*)d_in[5];
    const float* lnb = (const float*)d_in[6];
    float* out = (float*)d_out;

    char* ws = (char*)d_ws;
    size_t off = 0;
    auto alloc = [&](size_t bytes) -> void* {
        off = (off + 255) & ~(size_t)255;
        void* p = ws + off;
        off += bytes;
        return p;
    };
    float* imp  = (float*)alloc(sizeof(float) * BB * NN);
    int*   idx  = (int*)  alloc(sizeof(int)   * BB * KT);
    float* xsel = (float*)alloc(sizeof(float) * (size_t)BB * KT * CC);  // reused as 'attended'
    float* qsel = (float*)alloc(sizeof(float) * (size_t)BB * KT * CC);
    float* ksel = (float*)alloc(sizeof(float) * (size_t)BB * KT * CC);
    float* vbuf = (float*)alloc(sizeof(float) * (size_t)BB * CC * NN);
    float* vsel = (float*)alloc(sizeof(float) * (size_t)BB * KT * CC);
    float* attn = (float*)alloc(sizeof(float) * (size_t)BB * KT * KT);

    const long long sCN = (long long)CC * NN;
    const long long sKC = (long long)KT * CC;
    const long long sKK = (long long)KT * KT;

    importance_kernel<<<(BB * NN) / 256, 256, 0, stream>>>(x, imp);
    topk_kernel<<<BB, 1024, 0, stream>>>(imp, idx);

    gather_tokens<<<(BB * KT * CC) / 256, 256, 0, stream>>>(x, idx, xsel);

    // v = wv @ x over all tokens : (C x C) * (C x N)
    gemm_wmma_f32<false, 0><<<dim3(NN / 64, CC / 64, BB), 256, 0, stream>>>(
        wv, x, vbuf, nullptr, CC, CC, NN, NN, 0, sCN, sCN, 0);

    // q_sel/k_sel only over selected tokens : (KT x C) * w^T
    gemm_wmma_f32<true, 0><<<dim3(CC / 64, KT / 64, BB), 256, 0, stream>>>(
        xsel, wq, qsel, nullptr, CC, CC, CC, CC, sKC, 0, sKC, 0);
    gemm_wmma_f32<true, 0><<<dim3(CC / 64, KT / 64, BB), 256, 0, stream>>>(
        xsel, wk, ksel, nullptr, CC, CC, CC, CC, sKC, 0, sKC, 0);

    layernorm_rows<<<BB * KT, 256, 0, stream>>>(qsel, lnw, lnb);
    layernorm_rows<<<BB * KT, 256, 0, stream>>>(ksel, lnw, lnb);

    gather_tokens<<<(BB * KT * CC) / 256, 256, 0, stream>>>(vbuf, idx, vsel);

    // attn = qn @ kn^T * scale, nan->0, clamp
    gemm_wmma_f32<true, 1><<<dim3(KT / 64, KT / 64, BB), 256, 0, stream>>>(
        qsel, ksel, attn, nullptr, CC, CC, CC, KT, sKC, sKC, sKK, 0);

    softmax_rows<<<BB * KT, 256, 0, stream>>>(attn);

    // attended = attn @ v_sel -> reuse xsel
    gemm_wmma_f32<false, 0><<<dim3(CC / 64, KT / 64, BB), 256, 0, stream>>>(
        attn, vsel, xsel, nullptr, KT, KT, CC, CC, sKK, sKC, sKC, 0);

    scatter_tokens<<<(BB * KT * CC) / 256, 256, 0, stream>>>(idx, xsel, vbuf);

    // y = x + wo @ out_v
    gemm_wmma_f32<false, 2><<<dim3(NN / 64, CC / 64, BB), 256, 0, stream>>>(
        wo, vbuf, out, x, CC, CC, NN, NN, 0, sCN, sCN, sCN);
}